// MaskedMaxPool_60610578481786
// MI455X (gfx1250) — compile-verified
//
#include <hip/hip_runtime.h>

// ---------------------------------------------------------------------------
// PointNet++-style set abstraction for MI455X (gfx1250, wave32, WMMA, TDM).
//   xyz:      (8, 8192, 3)  f32
//   features: (8, 256, 8192) f32
// Outputs (concatenated in d_out):
//   new_xyz:      (8, 2048, 3)   f32   [FPS-sampled coords]
//   sub_features: (8, 256, 2048) f32   [masked max-pool over K=32 ball]
// ---------------------------------------------------------------------------

#define B_    8
#define N_    8192
#define C_    256
#define S_    2048     // NPOINT
#define K_    32       // nsample (== wave32 size, convenient)
#define R2_   0.04f    // radius^2

typedef __attribute__((ext_vector_type(2))) float        v2f;
typedef __attribute__((ext_vector_type(8))) float        v8f;
typedef __attribute__((ext_vector_type(4))) unsigned int v4u;
typedef __attribute__((ext_vector_type(4))) int          v4i;
typedef __attribute__((ext_vector_type(8))) int          v8i;

// ===========================================================================
// Kernel 1: farthest point sampling. One workgroup per batch, 1024 threads,
// 8 points/thread in registers; xyz mirrored in LDS for centroid broadcast.
// Matches jax scan semantics: emit `farthest` BEFORE the update; idx[0]=0.
// Argmax tie-break = smallest index (jnp.argmax picks first maximum).
// ===========================================================================
#define FPS_T 1024
#define FPS_P (N_ / FPS_T)   // 8 points per thread

__global__ void fps_kernel(const float* __restrict__ xyz,
                           float* __restrict__ new_xyz) {
  __shared__ float4 s_pts[N_];        // 128 KB
  __shared__ float  s_rv[32];
  __shared__ int    s_ri[32];
  __shared__ int    s_far;

  const int b    = blockIdx.x;
  const int tid  = threadIdx.x;
  const int lane = tid & 31;
  const int wave = tid >> 5;
  const float* xb = xyz + (size_t)b * N_ * 3;

  float px[FPS_P], py[FPS_P], pz[FPS_P], dd[FPS_P];
  #pragma unroll
  for (int k = 0; k < FPS_P; ++k) {
    const int n = tid + k * FPS_T;
    const float x = xb[n * 3 + 0];
    const float y = xb[n * 3 + 1];
    const float z = xb[n * 3 + 2];
    px[k] = x; py[k] = y; pz[k] = z;
    dd[k] = 1.0e10f;
    s_pts[n] = make_float4(x, y, z, 0.0f);
  }
  __syncthreads();

  int far = 0;
  float* outb = new_xyz + (size_t)b * S_ * 3;

  for (int i = 0; i < S_; ++i) {
    const float4 cen = s_pts[far];          // uniform LDS broadcast
    if (tid == 0) {
      outb[i * 3 + 0] = cen.x;
      outb[i * 3 + 1] = cen.y;
      outb[i * 3 + 2] = cen.z;
    }
    // update running min-distance, track local argmax (first-max)
    float bv = -1.0f; int bi = 0;
    #pragma unroll
    for (int k = 0; k < FPS_P; ++k) {
      const float dx = px[k] - cen.x;
      const float dy = py[k] - cen.y;
      const float dz = pz[k] - cen.z;
      const float dist = dx * dx + dy * dy + dz * dz;
      dd[k] = fminf(dd[k], dist);
      const int n = tid + k * FPS_T;
      if (dd[k] > bv) { bv = dd[k]; bi = n; }   // k ascending -> n ascending
    }
    // wave32 argmax reduce (min index on ties)
    #pragma unroll
    for (int off = 16; off >= 1; off >>= 1) {
      const float ov = __shfl_xor(bv, off, 32);
      const int   oi = __shfl_xor(bi, off, 32);
      if (ov > bv || (ov == bv && oi < bi)) { bv = ov; bi = oi; }
    }
    if (lane == 0) { s_rv[wave] = bv; s_ri[wave] = bi; }
    __syncthreads();
    if (wave == 0) {
      float bv2 = s_rv[lane];
      int   bi2 = s_ri[lane];
      #pragma unroll
      for (int off = 16; off >= 1; off >>= 1) {
        const float ov = __shfl_xor(bv2, off, 32);
        const int   oi = __shfl_xor(bi2, off, 32);
        if (ov > bv2 || (ov == bv2 && oi < bi2)) { bv2 = ov; bi2 = oi; }
      }
      if (lane == 0) s_far = bi2;
    }
    __syncthreads();
    far = s_far;
  }
}

// ===========================================================================
// Kernel 2: ball query via V_WMMA_F32_16X16X4_F32.
//   D[q][n] = A x B + C with A[q]=(-2x,-2y,-2z,|q|^2), B[:,n]=(x,y,z,1),
//   C[*]=|n|^2  ->  D = squared distance tile (16 queries x 16 points).
// VGPR layouts per CDNA5 ISA 7.12.2:
//   A (16x4 f32): lane<16 holds A[l][0],A[l][1]; lane>=16 holds A[l-16][2..3]
//   B (4x16 f32): v0 = rows K=0 (lanes 0-15) / K=2 (16-31); v1 = K=1 / K=3
//   D row r -> element (M = r + 8*(lane>>4), N = lane&15)
// Each wave owns 16 queries; points scanned in ascending tiles -> "first K
// in index order" via ballot + prefix popcount per half-wave.
// ===========================================================================
__global__ void ball_kernel(const float* __restrict__ xyz,
                            const float* __restrict__ new_xyz,
                            int* __restrict__ gidx) {
  __shared__ float4 s_pts[N_];        // (x, y, z, |n|^2), 128 KB

  const int b = blockIdx.y;
  const float* xb = xyz + (size_t)b * N_ * 3;
  for (int n = threadIdx.x; n < N_; n += 256) {
    const float x = xb[n * 3 + 0];
    const float y = xb[n * 3 + 1];
    const float z = xb[n * 3 + 2];
    s_pts[n] = make_float4(x, y, z, x * x + y * y + z * z);
  }
  __syncthreads();

  const int lane  = threadIdx.x & 31;
  const int wave  = threadIdx.x >> 5;
  const int tile  = blockIdx.x * 8 + wave;   // 0..127
  const int qbase = tile * 16;
  const int half  = lane >> 4;               // 0 or 1
  const int l15   = lane & 15;

  // A operand: query (qbase + l15)
  const float* qp = new_xyz + ((size_t)b * S_ + qbase + l15) * 3;
  const float qx = qp[0], qy = qp[1], qz = qp[2];
  const float qq = qx * qx + qy * qy + qz * qz;
  v2f A;
  A.x = half ? (-2.0f * qz) : (-2.0f * qx);
  A.y = half ? qq           : (-2.0f * qy);

  int cnt[8];
  int firstn[8];
  #pragma unroll
  for (int r = 0; r < 8; ++r) { cnt[r] = 0; firstn[r] = -1; }

  for (int nb = 0; nb < N_; nb += 16) {
    const float4 p = s_pts[nb + l15];
    v2f Bv;
    Bv.x = half ? p.z : p.x;
    Bv.y = half ? 1.0f : p.y;
    v8f Cv;
    #pragma unroll
    for (int r = 0; r < 8; ++r) Cv[r] = p.w;

    v8f D = __builtin_amdgcn_wmma_f32_16x16x4_f32(
        false, A, false, Bv, (short)0, Cv, false, false);

    #pragma unroll
    for (int r = 0; r < 8; ++r) {
      const bool pred = (D[r] <= R2_) && (cnt[r] < K_);
      const unsigned m  = (unsigned)__ballot(pred);
      const unsigned mh = (m >> (half * 16)) & 0xFFFFu;
      const int prefix = __popc(mh & ((1u << l15) - 1u));
      const int hits   = __popc(mh);
      const int slot   = cnt[r] + prefix;
      if (pred && slot < K_) {
        const int q = qbase + r + 8 * half;
        gidx[((size_t)b * S_ + q) * K_ + slot] = nb + l15;
      }
      if (firstn[r] < 0 && mh != 0u)
        firstn[r] = nb + (int)__builtin_ctz(mh);   // uniform per half-wave
      cnt[r] += hits;
    }
    bool mine = true;
    #pragma unroll
    for (int r = 0; r < 8; ++r) mine = mine && (cnt[r] >= K_);
    if (__all(mine)) break;                        // early exit, wave-uniform
  }

  // pad remaining slots with the first found index (reference semantics)
  #pragma unroll
  for (int qs = 0; qs < 16; ++qs) {
    int cq = __shfl(cnt[qs & 7],    (qs >> 3) * 16, 32);
    int fq = __shfl(firstn[qs & 7], (qs >> 3) * 16, 32);
    cq = min(cq, K_);
    if (fq < 0) fq = 0;                            // cannot happen (self-hit)
    if (lane >= cq)
      gidx[((size_t)b * S_ + qbase + qs) * K_ + lane] = fq;
  }
}

// ===========================================================================
// Kernel 3: gather + max-pool. Each workgroup stages CG full channel rows
// (CG*8192 f32 = 128 KB) in LDS, then waves gather K=32 neighbors per query
// from LDS and reduce with shfl-xor max. Staging uses the Tensor Data Mover
// (TENSOR_LOAD_TO_LDS): one 2D D# (tile 8192 x CG, dim0_stride 8192,
// data_size 4B) DMAs the whole block straight into LDS, tracked by TENSORcnt.
// Every feature byte is read from HBM exactly once (64 MB total).
// ===========================================================================
#define CG 4

__global__ void pool_kernel(const float* __restrict__ feat,
                            const int* __restrict__ gidx,
                            float* __restrict__ out) {
  __shared__ float s_rows[CG * N_];   // 128 KB

  const int b  = blockIdx.y;
  const int c0 = blockIdx.x * CG;
  const float* fb = feat + ((size_t)b * C_ + c0) * N_;

#if __has_builtin(__builtin_amdgcn_tensor_load_to_lds)
  if (threadIdx.x < 32) {   // one wave issues the DMA; TDM ignores EXEC
    const unsigned long long ga = (unsigned long long)fb;
    // Flat LDS pointers carry the LDS byte offset in addr[31:0] (ISA 10.2).
    const unsigned lds_addr = (unsigned)(unsigned long long)&s_rows[0];

    v4u g0;                                  // D# group 0 (ISA 8.3)
    g0[0] = 1u;                              // count=1, user descriptor
    g0[1] = lds_addr;                        // lds_addr[31:0]
    g0[2] = (unsigned)ga;                    // global_addr[31:0]
    g0[3] = (unsigned)((ga >> 32) & 0x1FFFFFFull) | (2u << 30); // addr[56:32], type=2

    v8i g1;                                  // D# group 1 (ISA 8.4)
    g1[0] = (2 << 16);                       // workgroup_mask=0, data_size=4B
    g1[1] = (int)((N_ & 0xFFFF) << 16);      // tensor_dim0[15:0]   (8192)
    g1[2] = (int)((N_ >> 16) | (CG << 16));  // tensor_dim0[31:16] | tensor_dim1[15:0]
    g1[3] = (int)((N_ & 0xFFFF) << 16);      // tensor_dim1[31:16]=0 | tile_dim0 (8192)
    g1[4] = CG;                              // tile_dim1=CG, tile_dim2=0
    g1[5] = N_;                              // tensor_dim0_stride[31:0] (8192)
    g1[6] = 0;                               // stride hi | tensor_dim1_stride lo
    g1[7] = 0;

    const v4i z4 = {0, 0, 0, 0};             // groups 2/3: tile_dim3/4 = 0
#if __clang_major__ >= 23
    const v8i z8 = {0, 0, 0, 0, 0, 0, 0, 0};
    __builtin_amdgcn_tensor_load_to_lds(g0, g1, z4, z4, z8, 0);
#else
    __builtin_amdgcn_tensor_load_to_lds(g0, g1, z4, z4, 0);
#endif
#if __has_builtin(__builtin_amdgcn_s_wait_tensorcnt)
    __builtin_amdgcn_s_wait_tensorcnt(0);
#else
    asm volatile("s_wait_tensorcnt 0x0" ::: "memory");
#endif
  }
#else
  for (int i = threadIdx.x; i < CG * N_ / 4; i += 1024)
    ((float4*)s_rows)[i] = ((const float4*)fb)[i];
#endif
  __syncthreads();

  const int lane = threadIdx.x & 31;
  const int wave = threadIdx.x >> 5;

  for (int q = wave; q < S_; q += 32) {
    const int idx = gidx[((size_t)b * S_ + q) * K_ + lane];
    #pragma unroll
    for (int c = 0; c < CG; ++c) {
      float v = s_rows[c * N_ + idx];
      #pragma unroll
      for (int off = 16; off >= 1; off >>= 1)
        v = fmaxf(v, __shfl_xor(v, off, 32));
      if (lane == 0)
        out[((size_t)b * C_ + c0 + c) * S_ + q] = v;
    }
  }
}

// ===========================================================================
extern "C" void kernel_launch(void* const* d_in, const int* in_sizes, int n_in,
                              void* d_out, int out_size, void* d_ws, size_t ws_size,
                              hipStream_t stream) {
  (void)in_sizes; (void)n_in; (void)out_size; (void)ws_size;
  const float* xyz  = (const float*)d_in[0];   // (8, 8192, 3)
  const float* feat = (const float*)d_in[1];   // (8, 256, 8192)

  float* new_xyz  = (float*)d_out;                       // 8*2048*3
  float* out_feat = (float*)d_out + (size_t)B_ * S_ * 3; // 8*256*2048
  int*   gidx     = (int*)d_ws;                          // 8*2048*32 ints (2 MB)

  fps_kernel <<<dim3(B_),      dim3(FPS_T), 0, stream>>>(xyz, new_xyz);
  ball_kernel<<<dim3(16, B_),  dim3(256),   0, stream>>>(xyz, new_xyz, gidx);
  pool_kernel<<<dim3(C_ / CG, B_), dim3(1024), 0, stream>>>(feat, gidx, out_feat);
}